// Encoder_71055938945027
// MI455X (gfx1250) — compile-verified
//
#include <hip/hip_runtime.h>
#include <hip/hip_bf16.h>
#include <stdint.h>

#define USER_NUM 200000
#define ITEM_NUM 100000
#define N_NODES (USER_NUM + ITEM_NUM)
#define EMB 64
#define N_EDGES 4800000
#define N_LAYERS 3

#ifndef __has_builtin
#define __has_builtin(x) 0
#endif

// ---- CDNA5 async load-to-LDS path (ASYNCcnt) -------------------------------
// Builtin signature (from hipcc diagnostic): (v4i as1*, v4i as3*, imm, imm)
#if __has_builtin(__builtin_amdgcn_global_load_async_to_lds_b128)
#define GCN_ASYNC 1
typedef int v4i __attribute__((ext_vector_type(4)));
typedef __attribute__((address_space(1))) v4i gv4i;   // global-memory int4
typedef __attribute__((address_space(3))) v4i lv4i;   // LDS int4
#else
#define GCN_ASYNC 0
#endif

#if __has_builtin(__builtin_amdgcn_s_wait_asynccnt)
#define WAIT_ASYNC(n) __builtin_amdgcn_s_wait_asynccnt(n)
#else
#define WAIT_ASYNC(n) asm volatile("s_wait_asynccnt " #n ::: "memory")
#endif

// Relaxed agent-scope f32 atomic add -> global_atomic_add_f32
// (no return, tracked with STOREcnt).
__device__ __forceinline__ void gatomic_fadd(float* p, float v) {
    __hip_atomic_fetch_add(p, v, __ATOMIC_RELAXED, __HIP_MEMORY_SCOPE_AGENT);
}

// ---- SpMM scatter: y[src] += val * x[dst] ----------------------------------
// wave32 mapping: lanes 0-15 -> edge 2b, lanes 16-31 -> edge 2b+1.
// Each lane owns 4 of the 64 dims (float4) => one contiguous 256B row per
// half-wave. Row gathers are software-pipelined through LDS with
// GLOBAL_LOAD_ASYNC_TO_LDS_B128 + s_wait_asynccnt double buffering
// (wave-private LDS region, so no workgroup barrier needed).
__global__ __launch_bounds__(256)
void spmm_scatter_kernel(const int* __restrict__ esrc,
                         const int* __restrict__ edst,
                         const float* __restrict__ eval,
                         const float* __restrict__ x,
                         float* __restrict__ y) {
    const int lane   = threadIdx.x & 31;
    const int eslot  = lane >> 4;    // which of the 2 edges this half-wave owns
    const int l16    = lane & 15;    // 0..15 -> dims [4*l16, 4*l16+3]
    const int gwave  = (blockIdx.x * blockDim.x + threadIdx.x) >> 5;
    const int nwaves = (gridDim.x * blockDim.x) >> 5;
    const int nb     = N_EDGES / 2;  // batches: 2 edges per wave per batch

#if GCN_ASYNC
    // 8 waves/block * 2 buffers * 2 edges * 64 floats = 8KB LDS
    __shared__ float lds[8 * 2 * 2 * EMB];
    float* wbase = &lds[(threadIdx.x >> 5) * (2 * 2 * EMB)];

    int b = gwave;
    int cur = 0;
    if (b < nb) {
        int e = 2 * b + eslot;
        int d = edst[e];
        __builtin_amdgcn_global_load_async_to_lds_b128(
            (gv4i*)(x + (size_t)d * EMB + l16 * 4),
            (lv4i*)(wbase + cur * (2 * EMB) + eslot * EMB + l16 * 4),
            0, 0);
    }
    for (; b < nb; b += nwaves) {
        const int nxt = b + nwaves;
        if (nxt < nb) {
            // prefetch edge streams two stages ahead
            int pe = 2 * (b + 2 * nwaves);
            if (pe < N_EDGES) {
                __builtin_prefetch(&edst[pe], 0, 1);
                __builtin_prefetch(&esrc[pe], 0, 1);
                __builtin_prefetch(&eval[pe], 0, 1);
            }
            int e = 2 * nxt + eslot;
            int d = edst[e];
            __builtin_amdgcn_global_load_async_to_lds_b128(
                (gv4i*)(x + (size_t)d * EMB + l16 * 4),
                (lv4i*)(wbase + (cur ^ 1) * (2 * EMB) + eslot * EMB + l16 * 4),
                0, 0);
            WAIT_ASYNC(1);   // current batch's gather is complete
        } else {
            WAIT_ASYNC(0);
        }
        const int e = 2 * b + eslot;
        const int s = esrc[e];
        const float v = eval[e];
        const float4 row =
            *(const float4*)(wbase + cur * (2 * EMB) + eslot * EMB + l16 * 4);
        float* yrow = y + (size_t)s * EMB + l16 * 4;
        gatomic_fadd(yrow + 0, v * row.x);
        gatomic_fadd(yrow + 1, v * row.y);
        gatomic_fadd(yrow + 2, v * row.z);
        gatomic_fadd(yrow + 3, v * row.w);
        cur ^= 1;
    }
#else
    for (int b = gwave; b < nb; b += nwaves) {
        const int e = 2 * b + eslot;
        const int d = edst[e];
        const int s = esrc[e];
        const float v = eval[e];
        const float4 row = *(const float4*)(x + (size_t)d * EMB + l16 * 4);
        float* yrow = y + (size_t)s * EMB + l16 * 4;
        gatomic_fadd(yrow + 0, v * row.x);
        gatomic_fadd(yrow + 1, v * row.y);
        gatomic_fadd(yrow + 2, v * row.z);
        gatomic_fadd(yrow + 3, v * row.w);
    }
#endif
}

// ---- ego0 = concat(user_emb, item_emb), vectorized float4 ------------------
__global__ void concat_kernel(const float* __restrict__ u,
                              const float* __restrict__ it,
                              float* __restrict__ out) {
    const int total4 = N_NODES * EMB / 4;
    int i = blockIdx.x * blockDim.x + threadIdx.x;
    if (i >= total4) return;
    const int usplit = USER_NUM * EMB / 4;
    float4 v;
    if (i < usplit) v = ((const float4*)u)[i];
    else            v = ((const float4*)it)[i - usplit];
    ((float4*)out)[i] = v;
}

__global__ void zero_kernel(float4* __restrict__ p, int n4) {
    int i = blockIdx.x * blockDim.x + threadIdx.x;
    if (i < n4) p[i] = make_float4(0.f, 0.f, 0.f, 0.f);
}

// acc = (first ? 0 : acc) + b * (1/N_LAYERS); optionally zero the dead ping
// buffer (next layer's scatter target) in the same streaming pass.
__global__ void accum_kernel(float* __restrict__ acc,
                             const float* __restrict__ b,
                             float* __restrict__ zbuf,   // may be nullptr
                             int first) {
    const int total4 = N_NODES * EMB / 4;
    int i = blockIdx.x * blockDim.x + threadIdx.x;
    if (i >= total4) return;
    const float s = 1.0f / (float)N_LAYERS;
    float4 v = ((const float4*)b)[i];
    float4 a;
    if (first) a = make_float4(0.f, 0.f, 0.f, 0.f);
    else       a = ((const float4*)acc)[i];
    a.x += v.x * s; a.y += v.y * s; a.z += v.z * s; a.w += v.w * s;
    ((float4*)acc)[i] = a;
    if (zbuf) ((float4*)zbuf)[i] = make_float4(0.f, 0.f, 0.f, 0.f);
}

extern "C" void kernel_launch(void* const* d_in, const int* in_sizes, int n_in,
                              void* d_out, int out_size, void* d_ws, size_t ws_size,
                              hipStream_t stream) {
    (void)in_sizes; (void)n_in; (void)out_size; (void)ws_size;
    const float* user_emb = (const float*)d_in[0];
    const float* item_emb = (const float*)d_in[1];
    const float* edge_val = (const float*)d_in[2];
    const int*   edge_src = (const int*)d_in[3];
    const int*   edge_dst = (const int*)d_in[4];

    float* A   = (float*)d_ws;                       // ego (ping)
    float* B   = A + (size_t)N_NODES * EMB;          // ego (pong)
    float* acc = (float*)d_out;                      // layer-mean accumulator

    const int n4 = N_NODES * EMB / 4;
    const int eltBlocks = (n4 + 255) / 256;

    concat_kernel<<<eltBlocks, 256, 0, stream>>>(user_emb, item_emb, A);
    zero_kernel<<<eltBlocks, 256, 0, stream>>>((float4*)B, n4);

    for (int layer = 0; layer < N_LAYERS; ++layer) {
        spmm_scatter_kernel<<<4096, 256, 0, stream>>>(edge_src, edge_dst,
                                                      edge_val, A, B);
        // Zero the now-dead source buffer A (next layer's scatter target)
        // in the same pass; last layer needs no zeroing.
        float* zbuf = (layer + 1 < N_LAYERS) ? A : nullptr;
        accum_kernel<<<eltBlocks, 256, 0, stream>>>(acc, B, zbuf,
                                                    layer == 0 ? 1 : 0);
        float* t = A; A = B; B = t;
    }
}